// AttentionWriter_63848983822690
// MI455X (gfx1250) — compile-verified
//
#include <hip/hip_runtime.h>
#include <hip/hip_bf16.h>
#include <math.h>

// AttentionWriter for MI455X (gfx1250, wave32).
//
// Decomposition:
//   attn_W = [Ws | Wg | Wm] column-wise (512 each).
//   Precompute (fp32 WMMA GEMMs, V_WMMA_F32_16X16X4_F32):
//     SQ[b,t,:] = states[b,t,:] @ Ws^T                  (32768 x 512, K=512)
//     SP[b,t,:] = states[b,t,:] @ Wm^T                  (32768 x 512, K=512)
//     GQ[b,:]   = global_trace[b,:] @ Wg^T + attn_b     (  256 x 512, K=512)
//     P0[b,s,:] = his_mem[b,s,:] @ Wm^T                 ( 3840 x 512, K=512)
//   Scan (1 workgroup per batch, 16 waves = 16 slots): incremental slot
//   projections P[s] in LDS, tanh+dot, softmax/gumbel/argmax, one-slot update.

typedef __attribute__((ext_vector_type(2))) float v2f;
typedef __attribute__((ext_vector_type(8))) float v8f;

#define TAU_F  1.0f
#define EPS_F  1e-20f

// Async global->LDS 16B copy (ASYNCcnt-tracked, no VGPR round trip).
__device__ __forceinline__ void async_copy_b128(unsigned lds_off,
                                                const void* gptr) {
    asm volatile("global_load_async_to_lds_b128 %0, %1, off"
                 :: "v"(lds_off), "v"(gptr) : "memory");
}
__device__ __forceinline__ void async_wait0() {
    asm volatile("s_wait_asynccnt 0x0" ::: "memory");
}

// ---------------------------------------------------------------------------
// GEMM: D[m][n] = sum_k A[m][k] * W[n][k]  (+ bias[n]),  K = 512.
// Block: 256 thr = 8 waves; block tile = 128 rows x 64 cols.
// Wave tile = 16 rows x 64 cols (4 accumulators sharing one A fetch -> 4x
// A reuse). W slice (64 x 512) staged in LDS via async b128 copies.
// ---------------------------------------------------------------------------
__global__ __launch_bounds__(256)
void gemm_wmma_f32(const float* __restrict__ A, const float* __restrict__ W,
                   const float* __restrict__ bias, float* __restrict__ D,
                   int ldw)
{
    __shared__ float Wl[64 * 516];            // padded stride: no bank conflicts
    const int tid  = threadIdx.x;
    const int lane = tid & 31;
    const int wv   = tid >> 5;                // 0..7
    const int half = lane >> 4;               // 0/1 (K split per ISA layout)
    const int r    = lane & 15;
    const int col0 = blockIdx.x * 64;
    const long row0 = (long)blockIdx.y * 128 + (long)wv * 16;

    // Stage the 64 x 512 slice of W (async, 16B per lane-transfer).
    // chunk c: row rr = c>>7, k-offset kk = (c&127)*4
    {
        const unsigned base = (unsigned)(size_t)&Wl[0];
#pragma unroll 4
        for (int c = tid; c < 64 * 128; c += 256) {
            const int rr = c >> 7, kk = (c & 127) << 2;
            async_copy_b128(base + (unsigned)(rr * 516 + kk) * 4u,
                            W + (long)(col0 + rr) * ldw + kk);
        }
        async_wait0();
    }
    __syncthreads();

    v8f c0, c1, c2, c3;
#pragma unroll
    for (int i = 0; i < 8; ++i) {
        c0[i] = bias ? bias[col0 + r]      : 0.0f;
        c1[i] = bias ? bias[col0 + 16 + r] : 0.0f;
        c2[i] = bias ? bias[col0 + 32 + r] : 0.0f;
        c3[i] = bias ? bias[col0 + 48 + r] : 0.0f;
    }

    const float* Arow = A + (row0 + r) * 512;       // A: lane r -> row M=r
    const float* W0 = Wl + (r)      * 516;          // B: lane r -> col N=r
    const float* W1 = Wl + (16 + r) * 516;
    const float* W2 = Wl + (32 + r) * 516;
    const float* W3 = Wl + (48 + r) * 516;

    // 16x16x4 f32 layout: v0 = K {0 | 2}, v1 = K {1 | 3} per half-wave.
#pragma unroll 4
    for (int k = 0; k < 512; k += 4) {
        const int ko = k + 2 * half;
        v2f a  = *(const v2f*)(Arow + ko);
        v2f b0 = *(const v2f*)(W0 + ko);
        v2f b1 = *(const v2f*)(W1 + ko);
        v2f b2 = *(const v2f*)(W2 + ko);
        v2f b3 = *(const v2f*)(W3 + ko);
        c0 = __builtin_amdgcn_wmma_f32_16x16x4_f32(false, a, false, b0,
                                                   (short)0, c0, false, false);
        c1 = __builtin_amdgcn_wmma_f32_16x16x4_f32(false, a, false, b1,
                                                   (short)0, c1, false, false);
        c2 = __builtin_amdgcn_wmma_f32_16x16x4_f32(false, a, false, b2,
                                                   (short)0, c2, false, false);
        c3 = __builtin_amdgcn_wmma_f32_16x16x4_f32(false, a, false, b3,
                                                   (short)0, c3, false, false);
    }

    // C/D layout: VGPR i -> M = i (lanes 0-15) / i+8 (lanes 16-31), N = lane&15
#pragma unroll
    for (int i = 0; i < 8; ++i) {
        float* drow = D + (row0 + i + 8 * half) * 512 + col0 + r;
        drow[0]  = c0[i];
        drow[16] = c1[i];
        drow[32] = c2[i];
        drow[48] = c3[i];
    }
}

// ---------------------------------------------------------------------------
// Per-row sum(|x|) for empty-slot detection: his_mem rows then states rows.
// One wave32 per 512-float row.
// ---------------------------------------------------------------------------
__global__ __launch_bounds__(256)
void row_abs_kernel(const float* __restrict__ his_mem,
                    const float* __restrict__ states,
                    float* __restrict__ hmAbs, float* __restrict__ stAbs)
{
    const int  lane = threadIdx.x & 31;
    long gw = (long)blockIdx.x * 8 + (threadIdx.x >> 5);
    const float* src;
    float* dst;
    if (gw < 3840) {                    // 256*15 his_mem rows
        src = his_mem + gw * 512; dst = hmAbs + gw;
    } else {
        gw -= 3840;
        if (gw >= 32768) return;        // 256*128 states rows
        src = states + gw * 512; dst = stAbs + gw;
    }
    float s = 0.0f;
    for (int c = 0; c < 16; ++c) s += fabsf(src[lane + 32 * c]);
#pragma unroll
    for (int o = 16; o; o >>= 1) s += __shfl_xor(s, o, 32);
    if (lane == 0) *dst = s;
}

// ---------------------------------------------------------------------------
// Sequential scan: one block (512 thr = 16 waves) per batch element.
// Wave s owns slot s (s==15 is the null slot, projection 0, always "empty").
// ---------------------------------------------------------------------------
__global__ __launch_bounds__(512)
void scan_kernel(const float* __restrict__ SQ, const float* __restrict__ GQ,
                 const float* __restrict__ P0, const float* __restrict__ SP,
                 const float* __restrict__ states,
                 const float* __restrict__ his_mem,
                 const float* __restrict__ smask,
                 const float* __restrict__ gumbel_u,
                 const float* __restrict__ v,
                 const float* __restrict__ stAbs,
                 const float* __restrict__ hmAbs,
                 float* __restrict__ hm_out, float* __restrict__ log_out)
{
    __shared__ float P[16][512];   // slot projections (mem @ Wm^T)
    __shared__ float sq[512];      // SQ[b,t,:] + GQ[b,:]  (bias folded in GQ)
    __shared__ float gq[512];
    __shared__ float vsh[512];
    __shared__ float E[16];
    __shared__ float emp[16];
    __shared__ int   lastw[16];
    __shared__ int   sel;

    const int  tid  = threadIdx.x;
    const int  lane = tid & 31;
    const int  wv   = tid >> 5;    // 0..15 == slot id
    const long b    = blockIdx.x;

    gq[tid]  = GQ[b * 512 + tid];
    vsh[tid] = v[tid];
    if (wv < 15) {
        for (int c = 0; c < 16; ++c)
            P[wv][lane + 32 * c] = P0[(b * 15 + wv) * 512 + lane + 32 * c];
    } else {
        for (int c = 0; c < 16; ++c) P[15][lane + 32 * c] = 0.0f;
    }
    if (tid < 16) {
        emp[tid] = (tid == 15) ? 1.0f
                 : ((hmAbs[b * 15 + tid] == 0.0f) ? 1.0f : 0.0f);
        lastw[tid] = -1;
    }
    __syncthreads();

    for (int t = 0; t < 128; ++t) {
        sq[tid] = SQ[(b * 128 + t) * 512 + tid] + gq[tid];
        __syncthreads();

        // energy_s = sum_j tanh(sq[j] + P[s][j]) * v[j]
        float p = 0.0f;
#pragma unroll
        for (int c = 0; c < 16; ++c) {
            const int j = lane + 32 * c;
            p += tanhf(sq[j] + P[wv][j]) * vsh[j];
        }
#pragma unroll
        for (int o = 16; o; o >>= 1) p += __shfl_xor(p, o, 32);
        if (lane == 0) E[wv] = p;
        __syncthreads();

        // wave 0, lanes 0..15: softmax(16) + empty*10 + gumbel + argmax
        if (wv == 0 && lane < 16) {
            const float e = E[lane];
            float m = e;
            for (int o = 8; o; o >>= 1) m = fmaxf(m, __shfl_xor(m, o, 16));
            float ex = __expf(e - m);
            float sum = ex;
            for (int o = 8; o; o >>= 1) sum += __shfl_xor(sum, o, 16);
            float aw = ex / sum + emp[lane] * 10.0f;

            float u = gumbel_u[(b * 128 + t) * 16 + lane];
            u = fminf(fmaxf(u, EPS_F), 1.0f);
            const float g = -__logf(-__logf(u) + EPS_F);
            float z = (aw + g) / TAU_F;

            int ki = lane;                       // argmax, first-max tie-break
            for (int o = 8; o; o >>= 1) {
                const float oz = __shfl_xor(z, o, 16);
                const int   oi = __shfl_xor(ki, o, 16);
                if (oz > z || (oz == z && oi < ki)) { z = oz; ki = oi; }
            }
            // forward value of straight-through gumbel-softmax == one-hot
            log_out[(b * 128 + t) * 16 + lane] = (lane == ki) ? 1.0f : 0.0f;
            if (lane == 0) {
                const float mk = smask[b * 128 + t];
                const int s2 = (ki < 15 && mk != 0.0f) ? ki : -1;
                sel = s2;
                if (s2 >= 0) {
                    lastw[s2] = t;
                    emp[s2] = (stAbs[b * 128 + t] == 0.0f) ? 1.0f : 0.0f;
                }
            }
        }
        __syncthreads();

        const int s2 = sel;                      // one-slot projection update
        if (s2 >= 0) P[s2][tid] = SP[(b * 128 + t) * 512 + tid];
        __syncthreads();
    }

    // Materialize hm_final from last-write indices.
    if (wv < 15) {
        const int lw = lastw[wv];
        const float* src = (lw >= 0) ? (states + (b * 128 + lw) * 512)
                                     : (his_mem + (b * 15 + wv) * 512);
        for (int c = 0; c < 16; ++c)
            hm_out[(b * 15 + wv) * 512 + lane + 32 * c] = src[lane + 32 * c];
    }
}

// ---------------------------------------------------------------------------
extern "C" void kernel_launch(void* const* d_in, const int* in_sizes, int n_in,
                              void* d_out, int out_size, void* d_ws, size_t ws_size,
                              hipStream_t stream)
{
    (void)in_sizes; (void)n_in; (void)out_size; (void)ws_size;

    const float* his_mem  = (const float*)d_in[0];   // [256,15,512]
    const float* states   = (const float*)d_in[1];   // [256,128,512]
    const float* smask    = (const float*)d_in[2];   // [256,128]
    const float* gtrace   = (const float*)d_in[3];   // [256,512]
    // d_in[4] = null_mem (all zeros) -- folded analytically
    const float* gumbel_u = (const float*)d_in[5];   // [256,128,16]
    const float* attn_W   = (const float*)d_in[6];   // [512,1536]
    const float* attn_b   = (const float*)d_in[7];   // [512]
    const float* v        = (const float*)d_in[8];   // [512]

    // workspace layout (floats): ~143 MB total
    float* ws    = (float*)d_ws;
    float* SQ    = ws;                        // 32768*512
    float* SP    = SQ + 32768l * 512;         // 32768*512
    float* GQ    = SP + 32768l * 512;         // 256*512
    float* P0    = GQ + 256l * 512;           // 3840*512
    float* stAbs = P0 + 3840l * 512;          // 32768
    float* hmAbs = stAbs + 32768;             // 3840

    float* hm_out  = (float*)d_out;                  // [256,15,512]
    float* log_out = hm_out + 256l * 15 * 512;       // [256,128,16]

    // Projection GEMMs (fp32 WMMA); column blocks of 64 -> grid.x = 8
    hipLaunchKernelGGL(gemm_wmma_f32, dim3(8, 256), dim3(256), 0, stream,
                       states, attn_W + 0,    nullptr, SQ, 1536);
    hipLaunchKernelGGL(gemm_wmma_f32, dim3(8, 256), dim3(256), 0, stream,
                       states, attn_W + 1024, nullptr, SP, 1536);
    hipLaunchKernelGGL(gemm_wmma_f32, dim3(8, 2),   dim3(256), 0, stream,
                       gtrace, attn_W + 512,  attn_b,  GQ, 1536);
    hipLaunchKernelGGL(gemm_wmma_f32, dim3(8, 30),  dim3(256), 0, stream,
                       his_mem, attn_W + 1024, nullptr, P0, 1536);

    hipLaunchKernelGGL(row_abs_kernel, dim3((3840 + 32768 + 7) / 8), dim3(256),
                       0, stream, his_mem, states, hmAbs, stAbs);

    hipLaunchKernelGGL(scan_kernel, dim3(256), dim3(512), 0, stream,
                       SQ, GQ, P0, SP, states, his_mem, smask, gumbel_u, v,
                       stAbs, hmAbs, hm_out, log_out);
}